// AnchorTarget_82308753261222
// MI455X (gfx1250) — compile-verified
//
#include <hip/hip_runtime.h>
#include <hip/hip_bf16.h>

#define GT_CAP   128
#define CAND_CAP 4096
#define NBIN     65536
#define POS_IOU  0.7f
#define NEG_IOU  0.3f

typedef unsigned int v4u __attribute__((ext_vector_type(4)));
typedef int          v8i __attribute__((ext_vector_type(8)));
typedef int          v4i __attribute__((ext_vector_type(4)));

// ---------------- JAX threefry2x32 (bit-exact) ----------------
__host__ __device__ static inline void tf2x32(unsigned k0, unsigned k1,
                                              unsigned x0, unsigned x1,
                                              unsigned& o0, unsigned& o1) {
  unsigned ks2 = 0x1BD11BDAu ^ k0 ^ k1;
  x0 += k0; x1 += k1;
#define TFR(r) { x0 += x1; x1 = (x1 << (r)) | (x1 >> (32 - (r))); x1 ^= x0; }
  TFR(13) TFR(15) TFR(26) TFR(6)   x0 += k1;  x1 += ks2 + 1u;
  TFR(17) TFR(29) TFR(16) TFR(24)  x0 += ks2; x1 += k0 + 2u;
  TFR(13) TFR(15) TFR(26) TFR(6)   x0 += k0;  x1 += k1 + 3u;
  TFR(17) TFR(29) TFR(16) TFR(24)  x0 += k1;  x1 += ks2 + 4u;
  TFR(13) TFR(15) TFR(26) TFR(6)   x0 += ks2; x1 += k0 + 5u;
#undef TFR
  o0 = x0; o1 = x1;
}

__device__ __forceinline__ float jax_uniform(unsigned k0, unsigned k1, int n, int N) {
  int half = N >> 1;
  unsigned c0 = (unsigned)((n < half) ? n : (n - half));
  unsigned c1 = c0 + (unsigned)half;
  unsigned o0, o1;
  tf2x32(k0, k1, c0, c1, o0, o1);
  unsigned bits = (n < half) ? o0 : o1;
  return __uint_as_float((bits >> 9) | 0x3f800000u) - 1.0f;
}

// order-preserving float<->uint for atomic max
__device__ __forceinline__ unsigned fflip(float f) {
  unsigned u = __float_as_uint(f);
  return (u & 0x80000000u) ? ~u : (u | 0x80000000u);
}
__device__ __forceinline__ float funflip(unsigned u) {
  u = (u & 0x80000000u) ? (u & 0x7fffffffu) : ~u;
  return __uint_as_float(u);
}

__device__ __forceinline__ float iou_one(float4 a, float4 b, float areaA, float areaB) {
  float yy1 = fmaxf(a.x, b.x), xx1 = fmaxf(a.y, b.y);
  float yy2 = fminf(a.z, b.z), xx2 = fminf(a.w, b.w);
  float ih = fmaxf(yy2 - yy1 + 1.0f, 0.0f);
  float iw = fmaxf(xx2 - xx1 + 1.0f, 0.0f);
  float inter = ih * iw;
  return inter / (areaA + areaB - inter);
}

// ---------------- TDM: async tensor load of GT table into LDS ----------------
__device__ __forceinline__ void load_gt_to_lds(void* lds_dst, const void* src, int nbytes) {
#if __has_builtin(__builtin_amdgcn_tensor_load_to_lds)
  if (threadIdx.x < 32) {   // wave 0 issues the DMA (EXEC ignored by TDM)
    unsigned long long ga = (unsigned long long)src;
    unsigned lds_off = (unsigned)(unsigned long long)lds_dst; // low 32 bits = LDS byte offset
    int ne = nbytes >> 2;   // f32 elements (data_size = 4B)
    v4u g0 = { 1u,                                  // count=1, user descriptor
               lds_off,                             // lds_addr
               (unsigned)(ga & 0xffffffffu),        // global_addr[31:0]
               ((unsigned)((ga >> 32) & 0x01ffffffu)) | 0x80000000u }; // addr[56:32] | type=2
    v8i g1 = { (int)0x00020000,                         // wg_mask=0, data_size=2 (4B)
               (int)(((unsigned)ne & 0xffffu) << 16),   // tensor_dim0 lo16
               (int)((((unsigned)ne >> 16) & 0xffffu) | (1u << 16)), // dim0 hi | tensor_dim1=1
               (int)(((unsigned)ne & 0xffffu) << 16),   // tile_dim0 = ne (1-D tile)
               0,                                       // tile_dim1=0, tile_dim2=0
               ne,                                      // tensor_dim0_stride
               0, 0 };
    v4i gz = { 0, 0, 0, 0 };
#if defined(__clang_major__) && (__clang_major__ >= 23)
    v8i g4 = { 0, 0, 0, 0, 0, 0, 0, 0 };
    __builtin_amdgcn_tensor_load_to_lds(g0, g1, gz, gz, g4, 0);
#else
    __builtin_amdgcn_tensor_load_to_lds(g0, g1, gz, gz, 0);
#endif
    __builtin_amdgcn_s_wait_tensorcnt(0);
  }
  __syncthreads();
#else
  int ne = nbytes >> 2;
  for (int i = (int)threadIdx.x; i < ne; i += (int)blockDim.x)
    ((float*)lds_dst)[i] = ((const float*)src)[i];
  __syncthreads();
#endif
}

// ---------------- kernels ----------------
__global__ void at_init(unsigned* hist_fg, unsigned* hist_bg, unsigned* gtmax, int* params) {
  int i = blockIdx.x * 256 + threadIdx.x;
  if (i < NBIN) { hist_fg[i] = 0u; hist_bg[i] = 0u; }
  if (i < GT_CAP) gtmax[i] = fflip(-1.0f);
  if (i < 64) params[i] = 0;
}

__global__ void at_main(const float4* __restrict__ anchors, const float4* __restrict__ gt,
                        const float* __restrict__ ishape,
                        float* __restrict__ uf, float* __restrict__ ub,
                        float* __restrict__ maxov, int* __restrict__ argmax,
                        unsigned* __restrict__ g_gtmax,
                        unsigned kf0, unsigned kf1, unsigned kb0, unsigned kb1,
                        int N, int G) {
  __shared__ float4   s_gt[GT_CAP];
  __shared__ float    s_area[GT_CAP];
  __shared__ unsigned s_gmx[GT_CAP];
  load_gt_to_lds(s_gt, gt, G * 16);
  int t = threadIdx.x;
  if (t < G) {
    float4 b = s_gt[t];
    s_area[t] = (b.z - b.x + 1.0f) * (b.w - b.y + 1.0f);
    s_gmx[t] = fflip(-1.0f);
  }
  __syncthreads();

  int n = blockIdx.x * 256 + t;
  bool act = n < N;
  float4 a = act ? anchors[n] : make_float4(0.f, 0.f, 0.f, 0.f);
  float h = ishape[0], w = ishape[1];
  bool valid = act && (a.x >= 0.f) && (a.y >= 0.f) && (a.z <= h - 1.0f) && (a.w <= w - 1.0f);
  float areaA = (a.z - a.x + 1.0f) * (a.w - a.y + 1.0f);

  float best = -3.402823466e38f;
  int bidx = 0;
  for (int g = 0; g < G; ++g) {
    float i = iou_one(a, s_gt[g], areaA, s_area[g]);
    if (i > best) { best = i; bidx = g; }       // first-max ties like jnp.argmax
    float v = valid ? i : -1.0f;                // ovm contribution
    for (int m = 16; m > 0; m >>= 1)            // wave32 max-reduce
      v = fmaxf(v, __shfl_xor(v, m));
    if ((t & 31) == 0) atomicMax(&s_gmx[g], fflip(v));
  }
  if (act) {
    maxov[n] = best;
    argmax[n] = bidx;
    uf[n] = jax_uniform(kf0, kf1, n, N);
    ub[n] = jax_uniform(kb0, kb1, n, N);
  }
  __syncthreads();
  if (t < G) atomicMax(&g_gtmax[t], s_gmx[t]);
}

__global__ void at_labels(const float4* __restrict__ anchors, const float4* __restrict__ gt,
                          const float* __restrict__ ishape,
                          const unsigned* __restrict__ g_gtmax,
                          const float* __restrict__ maxov,
                          const float* __restrict__ uf, const float* __restrict__ ub,
                          int* __restrict__ labels,
                          unsigned* __restrict__ hist_fg, unsigned* __restrict__ hist_bg,
                          int N, int G) {
  __shared__ float4 s_gt[GT_CAP];
  __shared__ float  s_area[GT_CAP];
  __shared__ float  s_gmxf[GT_CAP];
  load_gt_to_lds(s_gt, gt, G * 16);
  int t = threadIdx.x;
  if (t < G) {
    float4 b = s_gt[t];
    s_area[t] = (b.z - b.x + 1.0f) * (b.w - b.y + 1.0f);
    s_gmxf[t] = funflip(g_gtmax[t]);
  }
  __syncthreads();

  int n = blockIdx.x * 256 + t;
  if (n >= N) return;
  float4 a = anchors[n];
  float h = ishape[0], w = ishape[1];
  bool valid = (a.x >= 0.f) && (a.y >= 0.f) && (a.z <= h - 1.0f) && (a.w <= w - 1.0f);
  float areaA = (a.z - a.x + 1.0f) * (a.w - a.y + 1.0f);

  bool gbest = false;
  if (valid) {
    for (int g = 0; g < G; ++g) {
      float i = iou_one(a, s_gt[g], areaA, s_area[g]);  // identical fn -> identical bits
      gbest = gbest || (i == s_gmxf[g]);
    }
  }
  int l = -1;
  float mo = maxov[n];
  if (valid) {
    if (mo < NEG_IOU) l = 0;
    if (gbest)        l = 1;
    if (mo >= POS_IOU) l = 1;
  }
  labels[n] = l;
  if (l == 1)      atomicAdd(&hist_fg[__float_as_uint(uf[n]) >> 16], 1u);
  else if (l == 0) atomicAdd(&hist_bg[__float_as_uint(ub[n]) >> 16], 1u);
}

// params: 0 Tf, 1 C0f, 2 Tb, 3 C0b, 4 Mb, 5 inv_num(bits), 6 ccnt_fg, 7 ccnt_bg, 8 totfg, 9 totbg
__global__ void at_scan(const unsigned* __restrict__ hist_fg,
                        const unsigned* __restrict__ hist_bg, int* params) {
  __shared__ unsigned chunk[256];
  int t = threadIdx.x;
  for (int which = 0; which < 2; ++which) {
    const unsigned* hist = which ? hist_bg : hist_fg;
    int M = which ? params[4] : 128;
    unsigned local = 0;
    for (int k = 0; k < 256; ++k) local += hist[t * 256 + k];
    chunk[t] = local;
    __syncthreads();
    if (t == 0) {
      unsigned run = 0, pre = 0; int c = -1;
      for (int i = 0; i < 256; ++i) {
        unsigned cc = chunk[i];
        if (c < 0 && run + cc >= (unsigned)M) { c = i; pre = run; }
        run += cc;
      }
      int T, C0;
      if (c < 0) { T = 65536; C0 = (int)run; }     // fewer than M -> keep all
      else {
        unsigned p = pre; T = 65535; C0 = (int)pre;
        for (int k = 0; k < 256; ++k) {
          unsigned cb = hist[c * 256 + k];
          if (p + cb >= (unsigned)M) { T = c * 256 + k; C0 = (int)p; break; }
          p += cb;
        }
      }
      if (which == 0) {
        params[0] = T; params[1] = C0; params[8] = (int)run;
        int nf = (int)run < 128 ? (int)run : 128;
        params[4] = 256 - nf;                       // num_bg target
      } else {
        params[2] = T; params[3] = C0; params[9] = (int)run;
        int nf = params[8] < 128 ? params[8] : 128;
        int mb = params[4];
        int nb = (int)run < mb ? (int)run : mb;
        int ne = nf + nb;
        float inv = ne > 0 ? 1.0f / (float)ne : 0.0f;
        params[5] = __float_as_int(inv);
      }
    }
    __syncthreads();
  }
}

__global__ void at_apply(const float* __restrict__ uf, const float* __restrict__ ub,
                         int* params, int* labels,
                         unsigned* cand_fg, unsigned* cand_bg, int N) {
  int n = blockIdx.x * 256 + threadIdx.x;
  if (n >= N) return;
  int l = labels[n];
  if (l == 1) {
    unsigned bits = __float_as_uint(uf[n]);
    int key = (int)(bits >> 16), T = params[0];
    if (key > T) labels[n] = -1;
    else if (key == T) {
      int slot = atomicAdd(&params[6], 1);
      if (slot < CAND_CAP) { cand_fg[2 * slot] = bits; cand_fg[2 * slot + 1] = (unsigned)n; }
    }
  } else if (l == 0) {
    unsigned bits = __float_as_uint(ub[n]);
    int key = (int)(bits >> 16), T = params[2];
    if (key > T) labels[n] = -1;
    else if (key == T) {
      int slot = atomicAdd(&params[7], 1);
      if (slot < CAND_CAP) { cand_bg[2 * slot] = bits; cand_bg[2 * slot + 1] = (unsigned)n; }
    }
  }
}

__global__ void at_resolve(const unsigned* __restrict__ cand_fg,
                           const unsigned* __restrict__ cand_bg,
                           const int* __restrict__ params, int* labels) {
  int tid = blockIdx.x * 256 + threadIdx.x;
  int which = tid / CAND_CAP;
  int i = tid - which * CAND_CAP;
  if (which > 1) return;
  const unsigned* list = which ? cand_bg : cand_fg;
  int cnt = which ? params[7] : params[6];
  if (cnt > CAND_CAP) cnt = CAND_CAP;
  if (i >= cnt) return;
  int C0 = which ? params[3] : params[1];
  int M  = which ? params[4] : 128;
  unsigned u = list[2 * i], idx = list[2 * i + 1];
  int rank = 0;
  for (int j = 0; j < cnt; ++j) {
    unsigned uj = list[2 * j], ij = list[2 * j + 1];
    if (uj < u || (uj == u && ij < idx)) rank++;   // stable argsort tie-break
  }
  if (C0 + rank >= M) labels[idx] = -1;
}

__global__ void at_final(const float4* __restrict__ anchors, const float4* __restrict__ gt,
                         const float* __restrict__ ishape,
                         const int* __restrict__ labels, const int* __restrict__ argmax,
                         const int* __restrict__ params,
                         float* __restrict__ out, int N, int G) {
  __shared__ float4 s_gt[GT_CAP];
  load_gt_to_lds(s_gt, gt, G * 16);
  int n = blockIdx.x * 256 + threadIdx.x;
  if (n >= N) return;
  float4 a = anchors[n];
  float h = ishape[0], w = ishape[1];
  bool valid = (a.x >= 0.f) && (a.y >= 0.f) && (a.z <= h - 1.0f) && (a.w <= w - 1.0f);
  int l = labels[n];
  float inv = __int_as_float(params[5]);

  out[n] = (float)l;                               // labels (already -1 when invalid)

  float4 tgt = make_float4(0.f, 0.f, 0.f, 0.f);
  if (valid) {
    float4 g = s_gt[argmax[n]];
    float ah = a.z - a.x + 1.0f, aw = a.w - a.y + 1.0f;
    float ay = a.x + 0.5f * ah,  ax = a.y + 0.5f * aw;
    float gh = g.z - g.x + 1.0f, gw = g.w - g.y + 1.0f;
    float gy = g.x + 0.5f * gh,  gx = g.y + 0.5f * gw;
    tgt = make_float4((gy - ay) / ah, (gx - ax) / aw, logf(gh / ah), logf(gw / aw));
  }
  ((float4*)(out + (size_t)N))[n] = tgt;

  float iw = (l == 1) ? 1.0f : 0.0f;
  ((float4*)(out + (size_t)5 * N))[n] = make_float4(iw, iw, iw, iw);
  float ow = (l >= 0) ? inv : 0.0f;
  ((float4*)(out + (size_t)9 * N))[n] = make_float4(ow, ow, ow, ow);
}

// ---------------- host ----------------
extern "C" void kernel_launch(void* const* d_in, const int* in_sizes, int n_in,
                              void* d_out, int out_size, void* d_ws, size_t ws_size,
                              hipStream_t stream) {
  (void)n_in; (void)out_size; (void)ws_size;
  const float4* gt      = (const float4*)d_in[0];
  const float*  ishape  = (const float*)d_in[1];
  const float4* anchors = (const float4*)d_in[2];
  float* out = (float*)d_out;
  int G = in_sizes[0] / 4;
  int N = in_sizes[2] / 4;

  char* ws = (char*)d_ws;
  size_t o = 0;
  float*    uf      = (float*)(ws + o);    o += (size_t)N * 4;
  float*    ub      = (float*)(ws + o);    o += (size_t)N * 4;
  float*    maxov   = (float*)(ws + o);    o += (size_t)N * 4;
  int*      amax    = (int*)(ws + o);      o += (size_t)N * 4;
  int*      labels  = (int*)(ws + o);      o += (size_t)N * 4;
  unsigned* gtmax   = (unsigned*)(ws + o); o += GT_CAP * 4;
  unsigned* hist_fg = (unsigned*)(ws + o); o += (size_t)NBIN * 4;
  unsigned* hist_bg = (unsigned*)(ws + o); o += (size_t)NBIN * 4;
  int*      params  = (int*)(ws + o);      o += 256;
  unsigned* cand_fg = (unsigned*)(ws + o); o += (size_t)CAND_CAP * 8;
  unsigned* cand_bg = (unsigned*)(ws + o); o += (size_t)CAND_CAP * 8;

  // Replicate jax.random.key(42) -> split: counts [0,1,2,3], pairs (0,2) and (1,3)
  unsigned a0, b0, a1, b1;
  tf2x32(0u, 42u, 0u, 2u, a0, b0);
  tf2x32(0u, 42u, 1u, 3u, a1, b1);
  unsigned kf0 = a0, kf1 = a1;   // fg key
  unsigned kb0 = b0, kb1 = b1;   // bg key

  int nblk = (N + 255) / 256;
  dim3 blk(256);
  at_init  <<<(NBIN + 255) / 256, blk, 0, stream>>>(hist_fg, hist_bg, gtmax, params);
  at_main  <<<nblk, blk, 0, stream>>>(anchors, gt, ishape, uf, ub, maxov, amax, gtmax,
                                      kf0, kf1, kb0, kb1, N, G);
  at_labels<<<nblk, blk, 0, stream>>>(anchors, gt, ishape, gtmax, maxov, uf, ub,
                                      labels, hist_fg, hist_bg, N, G);
  at_scan  <<<1, blk, 0, stream>>>(hist_fg, hist_bg, params);
  at_apply <<<nblk, blk, 0, stream>>>(uf, ub, params, labels, cand_fg, cand_bg, N);
  at_resolve<<<(2 * CAND_CAP) / 256, blk, 0, stream>>>(cand_fg, cand_bg, params, labels);
  at_final <<<nblk, blk, 0, stream>>>(anchors, gt, ishape, labels, amax, params, out, N, G);
}